// GCNLayer_3332894622189
// MI455X (gfx1250) — compile-verified
//
#include <hip/hip_runtime.h>

// GCN layer for MI455X (gfx1250, wave32).
// N=8192 nodes, E=131072 edges, D_IN=D_OUT=64.
// Strategy: dense adjacency BITMAP for exact duplicate-edge dedup (matches
// reference .set() semantics), integer-atomic degree counts, deterministic
// per-row bitmap-scan gather (no float atomics), and the final projection
// h @ W^T + b done with V_WMMA_F32_16X16X4_F32 (fp32 matches reference math).

#define N_NODES 8192
#define D 64
#define MROW (N_NODES / 32)          // 256 mask words per adjacency row
#define MASK_WORDS (N_NODES * MROW)  // 2M words = 8 MB

typedef float v2f __attribute__((ext_vector_type(2)));
typedef float v8f __attribute__((ext_vector_type(8)));

// ---- 1. zero adjacency bitmap, deg = 1 (self loop from +I) ---------------
__global__ void gcn_init(unsigned* __restrict__ mask, int* __restrict__ deg) {
    int tid = blockIdx.x * blockDim.x + threadIdx.x;
    if (tid < MASK_WORDS) mask[tid] = 0u;
    if (tid < N_NODES)    deg[tid]  = 1;
}

// ---- 2. set bits; count degree only for NEWLY set bits (dedup) -----------
__global__ void gcn_edges(const long long* __restrict__ er,
                          const long long* __restrict__ ec,
                          unsigned* __restrict__ mask,
                          int* __restrict__ deg, int nE) {
    int e = blockIdx.x * blockDim.x + threadIdx.x;
    if (e >= nE) return;
    int r = (int)er[e];
    int c = (int)ec[e];
    unsigned bit = 1u << (c & 31);
    unsigned old = atomicOr(&mask[r * MROW + (c >> 5)], bit);
    if (!(old & bit)) atomicAdd(&deg[c], 1);   // integer: deterministic
}

// ---- 3. xs = d[:,None] * x ; store dinv ----------------------------------
__global__ void gcn_scale(const float* __restrict__ x, const int* __restrict__ deg,
                          float* __restrict__ xs, float* __restrict__ dinv) {
    int tid = blockIdx.x * blockDim.x + threadIdx.x;   // N*D threads
    int i = tid >> 6;
    float dv = rsqrtf((float)deg[i]);
    xs[tid] = x[tid] * dv;
    if ((tid & 63) == 0) dinv[i] = dv;
}

// ---- 4. h[r] = dinv[r] * (xs[r] + sum_{c in row r bitmap} xs[c]) ---------
// One 64-thread block per row; thread t owns feature t. Bitmap words are
// uniform across the block -> scalar loads; ascending-c order => deterministic.
__global__ void gcn_gather(const unsigned* __restrict__ mask,
                           const float* __restrict__ xs,
                           const float* __restrict__ dinv,
                           float* __restrict__ h) {
    int r = blockIdx.x;
    int t = threadIdx.x;                 // 0..63
    float acc = xs[r * D + t];           // self loop
    const unsigned* mrow = mask + r * MROW;
    for (int wi = 0; wi < MROW; ++wi) {
        unsigned w = mrow[wi];
        while (w) {
            int c = (wi << 5) + __builtin_ctz(w);
            acc += xs[c * D + t];        // 64 consecutive floats: coalesced
            w &= (w - 1);
        }
    }
    h[r * D + t] = dinv[r] * acc;
}

// ---- 5. out = h @ W^T + b via V_WMMA_F32_16X16X4_F32 ---------------------
// Wave computes a 16x64 tile: 4 N-tiles x 16 K-steps of 16x16x4 fp32 WMMA.
// A 16x4 layout: lanes 0-15 hold {K=0,K=1}, lanes 16-31 hold {K=2,K=3} for
// row M=lane%16 -> per-lane float2 at h[M][4s + 2*half].
// B 4x16 (=W^T tile): per-lane float2 at W[n0+lane%16][4s + 2*half].
// C/D: VGPR v holds (M = v + 8*half, N = n0 + lane%16).
__global__ void __launch_bounds__(256)
gcn_wmma(const float* __restrict__ h, const float* __restrict__ W,
         const float* __restrict__ bias, float* __restrict__ out) {
    int lane    = threadIdx.x & 31;
    int wave    = threadIdx.x >> 5;
    int rowbase = blockIdx.x * 128 + wave * 16;
    int half    = lane >> 4;
    int mrow    = lane & 15;
    int M       = rowbase + mrow;

    // Cache the full 16x64 A fragment (32 VGPRs/lane).
    v2f a[16];
    const v2f* hp = (const v2f*)(h + M * D);
    #pragma unroll
    for (int s = 0; s < 16; ++s) a[s] = hp[s * 2 + half];

    #pragma unroll
    for (int nt = 0; nt < 4; ++nt) {
        int n0 = nt * 16;
        const v2f* wp = (const v2f*)(W + (n0 + mrow) * D);
        v8f c = {0.f, 0.f, 0.f, 0.f, 0.f, 0.f, 0.f, 0.f};
        #pragma unroll
        for (int s = 0; s < 16; ++s) {
            v2f bfrag = wp[s * 2 + half];
            c = __builtin_amdgcn_wmma_f32_16x16x4_f32(
                    /*neg_a=*/false, a[s], /*neg_b=*/false, bfrag,
                    /*c_mod=*/(short)0, c, /*reuse_a=*/false, /*reuse_b=*/false);
        }
        float bb = bias[n0 + mrow];
        #pragma unroll
        for (int v = 0; v < 8; ++v) {
            int row = rowbase + v + 8 * half;
            out[row * D + n0 + mrow] = c[v] + bb;
        }
    }
}

extern "C" void kernel_launch(void* const* d_in, const int* in_sizes, int n_in,
                              void* d_out, int out_size, void* d_ws, size_t ws_size,
                              hipStream_t stream) {
    const float*     x  = (const float*)d_in[0];
    const long long* ei = (const long long*)d_in[1];   // int64 edge_index (2,E)
    const float*     W  = (const float*)d_in[2];
    const float*     b  = (const float*)d_in[3];
    float*           out = (float*)d_out;
    int nE = in_sizes[1] / 2;

    // Workspace layout (~12.1 MB total)
    char* wsp = (char*)d_ws;
    unsigned* mask = (unsigned*)wsp;                 size_t off = (size_t)MASK_WORDS * 4;
    int*   deg  = (int*)(wsp + off);                 off += (size_t)N_NODES * 4;
    float* dinv = (float*)(wsp + off);               off += (size_t)N_NODES * 4;
    float* xs   = (float*)(wsp + off);               off += (size_t)N_NODES * D * 4;
    float* h    = (float*)(wsp + off);

    gcn_init  <<<(MASK_WORDS + 255) / 256, 256, 0, stream>>>(mask, deg);
    gcn_edges <<<(nE + 255) / 256,         256, 0, stream>>>(ei, ei + nE, mask, deg, nE);
    gcn_scale <<<(N_NODES * D) / 256,      256, 0, stream>>>(x, deg, xs, dinv);
    gcn_gather<<<N_NODES,                   64, 0, stream>>>(mask, xs, dinv, h);
    gcn_wmma  <<<N_NODES / 128,            256, 0, stream>>>(h, W, b, out);
}